// KinematicL2_6459630813471
// MI455X (gfx1250) — compile-verified
//
#include <hip/hip_runtime.h>
#include <cstdint>

// ---------------------------------------------------------------------------
// Kinematic consistency loss for SE(3) trajectories.
// B blocks (one per trajectory), 256 threads each, double-buffered async
// global->LDS tile staging (CDNA5 GLOBAL_LOAD_ASYNC_TO_LDS_B128 + ASYNCcnt).
// ---------------------------------------------------------------------------

#define TILE   256              // tasks per tile
#define ROWS   (TILE + 1)       // rows of traj needed per tile (t and t+1)
#define ROW_F  12               // floats per trajectory row
#define BUF_F4 (ROWS * 3)       // 771 float4s per tile buffer
#define NLOAD  4                // ceil(BUF_F4 / 256) async loads per thread

struct M3 { float a00,a01,a02,a10,a11,a12,a20,a21,a22; };
struct V3 { float x,y,z; };

__device__ __forceinline__ M3 mmul(const M3& A, const M3& B) {
  M3 C;
  C.a00 = A.a00*B.a00 + A.a01*B.a10 + A.a02*B.a20;
  C.a01 = A.a00*B.a01 + A.a01*B.a11 + A.a02*B.a21;
  C.a02 = A.a00*B.a02 + A.a01*B.a12 + A.a02*B.a22;
  C.a10 = A.a10*B.a00 + A.a11*B.a10 + A.a12*B.a20;
  C.a11 = A.a10*B.a01 + A.a11*B.a11 + A.a12*B.a21;
  C.a12 = A.a10*B.a02 + A.a11*B.a12 + A.a12*B.a22;
  C.a20 = A.a20*B.a00 + A.a21*B.a10 + A.a22*B.a20;
  C.a21 = A.a20*B.a01 + A.a21*B.a11 + A.a22*B.a21;
  C.a22 = A.a20*B.a02 + A.a21*B.a12 + A.a22*B.a22;
  return C;
}

// A * B^T
__device__ __forceinline__ M3 mmulT(const M3& A, const M3& B) {
  M3 C;
  C.a00 = A.a00*B.a00 + A.a01*B.a01 + A.a02*B.a02;
  C.a01 = A.a00*B.a10 + A.a01*B.a11 + A.a02*B.a12;
  C.a02 = A.a00*B.a20 + A.a01*B.a21 + A.a02*B.a22;
  C.a10 = A.a10*B.a00 + A.a11*B.a01 + A.a12*B.a02;
  C.a11 = A.a10*B.a10 + A.a11*B.a11 + A.a12*B.a12;
  C.a12 = A.a10*B.a20 + A.a11*B.a21 + A.a12*B.a22;
  C.a20 = A.a20*B.a00 + A.a21*B.a01 + A.a22*B.a02;
  C.a21 = A.a20*B.a10 + A.a21*B.a11 + A.a22*B.a12;
  C.a22 = A.a20*B.a20 + A.a21*B.a21 + A.a22*B.a22;
  return C;
}

__device__ __forceinline__ V3 mv(const M3& A, const V3& v) {
  V3 r;
  r.x = A.a00*v.x + A.a01*v.y + A.a02*v.z;
  r.y = A.a10*v.x + A.a11*v.y + A.a12*v.z;
  r.z = A.a20*v.x + A.a21*v.y + A.a22*v.z;
  return r;
}

// se3_exp: xi = [rho(3), phi(3)]; R = I + A K + B K^2 ; t = (I + B K + C K^2) rho
// Uses K^2 = phi phi^T - |phi|^2 I to build matrices directly (branchless).
__device__ __forceinline__ void se3_exp(float rx, float ry, float rz,
                                        float px, float py, float pz,
                                        M3& R, V3& t) {
  float th2 = px*px + py*py + pz*pz;
  float th  = __builtin_sqrtf(th2);
  bool  small = th < 1e-4f;
  float ths = small ? 1.0f : th;
  float s, c;
  __sincosf(ths, &s, &c);
  float it  = __builtin_amdgcn_rcpf(ths);
  float it2 = it * it;
  float A = small ? 1.0f - th2*(1.0f/6.0f)              : s * it;
  float B = small ? 0.5f - th2*(1.0f/24.0f)             : (1.0f - c) * it2;
  float C = small ? (1.0f/6.0f) - th2*(1.0f/120.0f)     : (ths - s) * it2 * it;

  float Ax=A*px, Ay=A*py, Az=A*pz;
  float Bxx=B*px*px, Byy=B*py*py, Bzz=B*pz*pz;
  float Bxy=B*px*py, Bxz=B*px*pz, Byz=B*py*pz;
  R.a00 = 1.0f - Byy - Bzz; R.a01 = Bxy - Az;        R.a02 = Bxz + Ay;
  R.a10 = Bxy + Az;         R.a11 = 1.0f - Bxx - Bzz; R.a12 = Byz - Ax;
  R.a20 = Bxz - Ay;         R.a21 = Byz + Ax;        R.a22 = 1.0f - Bxx - Byy;

  float Bx=B*px, By=B*py, Bz=B*pz;
  float Cxx=C*px*px, Cyy=C*py*py, Czz=C*pz*pz;
  float Cxy=C*px*py, Cxz=C*px*pz, Cyz=C*py*pz;
  t.x = (1.0f - Cyy - Czz)*rx + (Cxy - Bz)*ry + (Cxz + By)*rz;
  t.y = (Cxy + Bz)*rx + (1.0f - Cxx - Czz)*ry + (Cyz - Bx)*rz;
  t.z = (Cxz - By)*rx + (Cyz + Bx)*ry + (1.0f - Cxx - Cyy)*rz;
}

// se3_log(R,t) -> |rho|^2 + |phi|^2
__device__ __forceinline__ float se3_log_sq(const M3& R, const V3& t) {
  float tr  = R.a00 + R.a11 + R.a22;
  float cth = fminf(1.0f, fmaxf(-1.0f, (tr - 1.0f) * 0.5f));
  float th  = acosf(cth);
  float th2 = th * th;
  bool  small = th < 1e-4f;
  float ths = small ? 1.0f : th;
  float s, c;
  __sincosf(ths, &s, &c);
  float it  = __builtin_amdgcn_rcpf(ths);
  float it2 = it * it;

  float wx = 0.5f*(R.a21 - R.a12);
  float wy = 0.5f*(R.a02 - R.a20);
  float wz = 0.5f*(R.a10 - R.a01);
  float fac = small ? 1.0f + th2*(1.0f/6.0f) : ths * __builtin_amdgcn_rcpf(s);
  float px = fac*wx, py = fac*wy, pz = fac*wz;

  float A = small ? 1.0f - th2*(1.0f/6.0f)  : s * it;
  float B = small ? 0.5f - th2*(1.0f/24.0f) : (1.0f - c) * it2;
  float D = small ? (1.0f/12.0f) + th2*(1.0f/720.0f)
                  : (1.0f - A * 0.5f * __builtin_amdgcn_rcpf(B)) * it2;

  // Vinv = I - 0.5 K + D K^2  (K = hat(phi))
  float Dxx=D*px*px, Dyy=D*py*py, Dzz=D*pz*pz;
  float Dxy=D*px*py, Dxz=D*px*pz, Dyz=D*py*pz;
  float hx=0.5f*px, hy=0.5f*py, hz=0.5f*pz;
  float rx = (1.0f - Dyy - Dzz)*t.x + (Dxy + hz)*t.y + (Dxz - hy)*t.z;
  float ry = (Dxy - hz)*t.x + (1.0f - Dxx - Dzz)*t.y + (Dyz + hx)*t.z;
  float rz = (Dxz + hy)*t.x + (Dyz - hx)*t.y + (1.0f - Dxx - Dyy)*t.z;
  return rx*rx + ry*ry + rz*rz + px*px + py*py + pz*pz;
}

__device__ __forceinline__ float task_loss(const float* __restrict__ x0,
                                           const float* __restrict__ x1) {
  const float hdt = 0.05f; // DT/2
  M3 HR, FR, HdR, BR;
  V3 Ht, Ft, Hdt, Bt;
  se3_exp(x0[0], x0[1], x0[2], x0[3], x0[4], x0[5], HR, Ht);
  se3_exp(x0[6]*hdt,  x0[7]*hdt,  x0[8]*hdt,  x0[9]*hdt,  x0[10]*hdt,  x0[11]*hdt,  FR, Ft);
  se3_exp(x1[0], x1[1], x1[2], x1[3], x1[4], x1[5], HdR, Hdt);
  se3_exp(-x1[6]*hdt, -x1[7]*hdt, -x1[8]*hdt, -x1[9]*hdt, -x1[10]*hdt, -x1[11]*hdt, BR, Bt);

  M3 fR = mmul(FR, HR);
  V3 fv = mv(FR, Ht);  V3 ft = { fv.x + Ft.x, fv.y + Ft.y, fv.z + Ft.z };
  M3 bR = mmul(BR, HdR);
  V3 bv = mv(BR, Hdt); V3 bt = { bv.x + Bt.x, bv.y + Bt.y, bv.z + Bt.z };

  M3 dR = mmulT(fR, bR);             // fwd_R * bwd_R^T
  V3 db = mv(dR, bt);
  V3 dt = { ft.x - db.x, ft.y - db.y, ft.z - db.z }; // fwd_t - dR*bwd_t
  return se3_log_sq(dR, dt);
}

__global__ __launch_bounds__(256)
void kin_block_kernel(const float* __restrict__ traj,
                      const float* __restrict__ t_weights,
                      const float* __restrict__ k_weights,
                      const int*   __restrict__ kidx,
                      float* __restrict__ out,
                      int H) {
  __shared__ __align__(16) float buf[2][ROWS * ROW_F]; // 2 x 12336 B
  __shared__ float red[256];

  const int b   = blockIdx.x;
  const int tid = threadIdx.x;
  const int T   = H - 1;
  const float* base = traj + (size_t)b * (size_t)H * ROW_F;
  const long maxf4  = ((long)H * ROW_F) / 4 - 1;
  const int ntiles  = (T + TILE - 1) / TILE;

  // Issue one tile's worth of async DMA copies (global -> LDS).
  // Uniform NLOAD instructions per wave (clamped addresses) so every wave's
  // ASYNCcnt advances identically: exact double-buffer bookkeeping.
  auto issue_tile = [&](int ti, int sel) {
    long f4base = (long)ti * TILE * 3;
#pragma unroll
    for (int kk = 0; kk < NLOAD; ++kk) {
      int j = tid + kk * 256;
      if (j >= BUF_F4) j = BUF_F4 - 1;          // duplicate tail load (harmless)
      long gf4 = f4base + j;
      if (gf4 > maxf4) gf4 = maxf4;             // clamp inside trajectory b
      uint32_t lds_off = (uint32_t)(uintptr_t)(&buf[sel][j * 4]);
      uint64_t gaddr   = (uint64_t)(uintptr_t)(base + gf4 * 4);
      asm volatile("global_load_async_to_lds_b128 %0, %1, off"
                   :: "v"(lds_off), "v"(gaddr) : "memory");
    }
  };

  float acc = 0.0f;
  issue_tile(0, 0);
  for (int i = 0; i < ntiles; ++i) {
    const int sel = i & 1;
    if (i + 1 < ntiles) {
      issue_tile(i + 1, sel ^ 1);               // prefetch next tile via DMA
      asm volatile("s_wait_asynccnt 4" ::: "memory"); // tile i done, i+1 in flight
    } else {
      asm volatile("s_wait_asynccnt 0" ::: "memory");
    }
    __syncthreads();                            // all waves' DMA visible in LDS
    const int t = i * TILE + tid;
    if (t < T) {
      const float* x0 = &buf[sel][tid * ROW_F];
      acc += task_loss(x0, x0 + ROW_F) * t_weights[t];
    }
    __syncthreads();                            // compute done before buffer reuse
  }

  red[tid] = acc;
  __syncthreads();
  for (int s = 128; s > 0; s >>= 1) {
    if (tid < s) red[tid] += red[tid + s];
    __syncthreads();
  }
  if (tid == 0) {
    float w = (red[0] / (float)T) * k_weights[kidx[b]];
    out[1 + b] = w;
  }
}

__global__ __launch_bounds__(1024)
void finalize_kernel(float* __restrict__ out, const int* __restrict__ kidx, int B) {
  __shared__ float ssum[1024];
  __shared__ float smax[1024];
  __shared__ int   sarg[1024];
  int tid = threadIdx.x;
  float v = (tid < B) ? out[1 + tid] : 0.0f;
  ssum[tid] = (tid < B) ? v : 0.0f;
  smax[tid] = (tid < B) ? v : -INFINITY;
  sarg[tid] = tid;
  __syncthreads();
  for (int s = 512; s > 0; s >>= 1) {
    if (tid < s) {
      ssum[tid] += ssum[tid + s];
      float o = smax[tid + s]; int oi = sarg[tid + s];
      // first-occurrence argmax (lower index wins ties), matching jnp.argmax
      if (o > smax[tid] || (o == smax[tid] && oi < sarg[tid])) {
        smax[tid] = o; sarg[tid] = oi;
      }
    }
    __syncthreads();
  }
  if (tid == 0) {
    out[0]     = ssum[0] / (float)B;   // weighted.mean()
    out[1 + B] = smax[0];              // weighted[idx]
    out[2 + B] = (float)kidx[sarg[0]]; // k[idx]
  }
}

extern "C" void kernel_launch(void* const* d_in, const int* in_sizes, int n_in,
                              void* d_out, int out_size, void* d_ws, size_t ws_size,
                              hipStream_t stream) {
  const float* traj = (const float*)d_in[0];
  const float* tw   = (const float*)d_in[1];
  const float* kw   = (const float*)d_in[2];
  const int*   kk   = (const int*)d_in[3];
  float* out = (float*)d_out;

  const int B = in_sizes[3];                 // 1024
  const int H = in_sizes[0] / (B * ROW_F);   // 2048

  kin_block_kernel<<<B, 256, 0, stream>>>(traj, tw, kw, kk, out, H);
  finalize_kernel<<<1, 1024, 0, stream>>>(out, kk, B);
}